// GAT_19335942766937
// MI455X (gfx1250) — compile-verified
//
#include <hip/hip_runtime.h>

typedef __attribute__((ext_vector_type(2))) float v2f;
typedef __attribute__((ext_vector_type(8))) float v8f;

#define HH 4          // heads
#define CC 32         // channels per head
#define FDIM 128      // H*C
#define NEG_SLOPE 0.2f

// ---------------------------------------------------------------------------
// FP32 WMMA GEMM: D[M,NCOLS] = act(A[M,KDIM] * B[KDIM,NCOLS] + bias)
// - One wave computes a 16-row strip across all NCOLS (A fragment reused
//   across NT column tiles). Requires M % 16 == 0 (true: N=100000).
// - B (the weight matrix) is staged into LDS per block in 64-row chunks;
//   LDS row pitch NCOLS+8 makes the two half-wave K-row reads bank-disjoint.
//
// V_WMMA_F32_16X16X4_F32 operand layouts (ISA 7.12.2):
//   A 16x4 : lanes 0-15 hold M=lane, VGPR0=K0,VGPR1=K1 ; lanes16-31 K2,K3
//   B 4x16 : lanes 0-15 hold N=lane, VGPR0=K0,VGPR1=K1 ; lanes16-31 K2,K3
//   C 16x16: VGPR r -> M = r + 8*(lane>=16), N = lane&15
// ---------------------------------------------------------------------------
template <int KDIM, int NCOLS>
__global__ __launch_bounds__(256) void gemm_wmma_f32(
    const float* __restrict__ A, const float* __restrict__ B,
    float* __restrict__ D, int M, const float* __restrict__ bias,
    int do_relu) {
  constexpr int NT = NCOLS / 16;
  constexpr int KC = (KDIM > 64) ? 64 : KDIM;   // K-chunk staged in LDS
  constexpr int PITCH = NCOLS + 8;              // words; bank-conflict pad
  __shared__ __align__(16) float Bs[KC * PITCH];

  const int wave = threadIdx.x >> 5;
  const int lane = threadIdx.x & 31;
  const int half = lane >> 4;   // 0: K{0,1}, 1: K{2,3}
  const int m16  = lane & 15;
  const long row0 = ((long)blockIdx.x * 8 + wave) * 16;
  const bool active = (row0 < M);

  v8f acc[NT];
#pragma unroll
  for (int t = 0; t < NT; ++t) acc[t] = (v8f){};

  const float* arow = A + (row0 + m16) * KDIM + 2 * half;

  for (int k0 = 0; k0 < KDIM; k0 += KC) {
    // ---- cooperative stage of B rows [k0, k0+KC) into LDS (float4) ----
    for (int i = threadIdx.x; i < KC * (NCOLS / 4); i += 256) {
      int r  = i / (NCOLS / 4);
      int c4 = i % (NCOLS / 4);
      float4 v = *(const float4*)(B + (size_t)(k0 + r) * NCOLS + c4 * 4);
      *(float4*)(Bs + r * PITCH + c4 * 4) = v;
    }
    __syncthreads();

    if (active) {
      for (int k = 0; k < KC; k += 4) {
        v2f a;
        a.x = arow[k0 + k + 0];
        a.y = arow[k0 + k + 1];
        const float* bp = Bs + (k + 2 * half) * PITCH + m16;
#pragma unroll
        for (int t = 0; t < NT; ++t) {
          v2f b;
          b.x = bp[t * 16];
          b.y = bp[t * 16 + PITCH];
          acc[t] = __builtin_amdgcn_wmma_f32_16x16x4_f32(
              false, a, false, b, (short)0, acc[t], false, false);
        }
      }
    }
    __syncthreads();
  }

  if (!active) return;
#pragma unroll
  for (int t = 0; t < NT; ++t) {
    float bv = bias ? bias[t * 16 + m16] : 0.0f;
#pragma unroll
    for (int r = 0; r < 8; ++r) {
      float v = acc[t][r] + bv;
      if (do_relu) v = fmaxf(v, 0.0f);
      D[(row0 + r + 8 * half) * NCOLS + t * 16 + m16] = v;
    }
  }
}

// ---------------------------------------------------------------------------
// Per-(node,head) attention coefficients: a = <h[n,hh,:], att[hh,:]>
// ---------------------------------------------------------------------------
__global__ void att_coef_kernel(const float* __restrict__ h,
                                const float* __restrict__ att_src,
                                const float* __restrict__ att_dst,
                                float* __restrict__ a_src,
                                float* __restrict__ a_dst, int NH) {
  int idx = blockIdx.x * blockDim.x + threadIdx.x;
  if (idx >= NH) return;
  int n = idx >> 2, hh = idx & 3;
  const float* hp = h + (long)n * FDIM + hh * CC;
  const float* ws = att_src + hh * CC;
  const float* wd = att_dst + hh * CC;
  float s = 0.f, d = 0.f;
#pragma unroll
  for (int c = 0; c < CC; ++c) {
    float v = hp[c];
    s += v * ws[c];
    d += v * wd[c];
  }
  a_src[idx] = s;
  a_dst[idx] = d;
}

// float atomic max via sign-split int/uint trick (m initialized to 0xFFFFFFFF)
__device__ inline void atomicMaxF(float* addr, float val) {
  if (val >= 0.0f)
    atomicMax((int*)addr, __float_as_int(val));
  else
    atomicMin((unsigned int*)addr, __float_as_uint(val));
}

__device__ inline float edge_logit(const float* a_src, const float* a_dst,
                                   int s, int d, int hh) {
  float e = a_src[s * HH + hh] + a_dst[d * HH + hh];
  return e > 0.f ? e : NEG_SLOPE * e;
}

__global__ void edge_max_kernel(const int* __restrict__ src,
                                const int* __restrict__ dst,
                                const float* __restrict__ a_src,
                                const float* __restrict__ a_dst,
                                float* __restrict__ m, int E) {
  int i = blockIdx.x * blockDim.x + threadIdx.x;
  if (i >= E) return;
  int s = src[i], d = dst[i];
#pragma unroll
  for (int hh = 0; hh < HH; ++hh)
    atomicMaxF(&m[d * HH + hh], edge_logit(a_src, a_dst, s, d, hh));
}

__global__ void edge_sum_kernel(const int* __restrict__ src,
                                const int* __restrict__ dst,
                                const float* __restrict__ a_src,
                                const float* __restrict__ a_dst,
                                const float* __restrict__ m,
                                float* __restrict__ z, int E) {
  int i = blockIdx.x * blockDim.x + threadIdx.x;
  if (i >= E) return;
  int s = src[i], d = dst[i];
#pragma unroll
  for (int hh = 0; hh < HH; ++hh) {
    float e = edge_logit(a_src, a_dst, s, d, hh);
    atomicAdd(&z[d * HH + hh], __expf(e - m[d * HH + hh]));
  }
}

// One wave32 per edge: lane gathers float4 of h[src] (128 floats total),
// scales by the per-head softmax weight, scatters with f32 atomics.
__global__ __launch_bounds__(256) void edge_scatter_kernel(
    const int* __restrict__ src, const int* __restrict__ dst,
    const float* __restrict__ a_src, const float* __restrict__ a_dst,
    const float* __restrict__ m, const float* __restrict__ z,
    const float* __restrict__ h, float* __restrict__ agg, int E) {
  int wid = (blockIdx.x * blockDim.x + threadIdx.x) >> 5;
  int lane = threadIdx.x & 31;
  if (wid >= E) return;
  int s = src[wid], d = dst[wid];
  int hh = lane >> 3;  // 8 lanes * float4 = 32 channels per head
  float e = edge_logit(a_src, a_dst, s, d, hh);
  float w = __expf(e - m[d * HH + hh]) / z[d * HH + hh];
  const float4 hv = ((const float4*)(h + (long)s * FDIM))[lane];
  float* o = agg + (long)d * FDIM + lane * 4;
  atomicAdd(o + 0, w * hv.x);
  atomicAdd(o + 1, w * hv.y);
  atomicAdd(o + 2, w * hv.z);
  atomicAdd(o + 3, w * hv.w);
}

// In-place x[i] = relu(x[i] + bias[i & colmask]); cols is a power of two.
__global__ void bias_relu_kernel(float* __restrict__ x,
                                 const float* __restrict__ b, int colmask,
                                 long total) {
  long i = (long)blockIdx.x * blockDim.x + threadIdx.x;
  if (i >= total) return;
  x[i] = fmaxf(x[i] + b[i & colmask], 0.0f);
}

// ---------------------------------------------------------------------------
extern "C" void kernel_launch(void* const* d_in, const int* in_sizes, int n_in,
                              void* d_out, int out_size, void* d_ws,
                              size_t ws_size, hipStream_t stream) {
  const float* x        = (const float*)d_in[0];
  const int*   eidx     = (const int*)d_in[1];
  const float* W0       = (const float*)d_in[2];
  const float* att_src0 = (const float*)d_in[3];
  const float* att_dst0 = (const float*)d_in[4];
  const float* b0       = (const float*)d_in[5];
  const float* W1       = (const float*)d_in[6];
  const float* att_src1 = (const float*)d_in[7];
  const float* att_dst1 = (const float*)d_in[8];
  const float* b1       = (const float*)d_in[9];
  const float* Wm1      = (const float*)d_in[10];
  const float* bm1      = (const float*)d_in[11];
  const float* Wm2      = (const float*)d_in[12];
  const float* bm2      = (const float*)d_in[13];
  float* out = (float*)d_out;

  const int N = in_sizes[0] / FDIM;
  const int E = in_sizes[1] / 2;
  const int* src = eidx;
  const int* dst = eidx + E;

  // Carve workspace (256B aligned chunks).
  size_t off = 0;
  auto carve = [&](size_t nfl) {
    float* p = (float*)((char*)d_ws + off);
    off += ((nfl * sizeof(float) + 255) / 256) * 256;
    return p;
  };
  float* hbuf = carve((size_t)N * FDIM);   // h of current layer / MLP hidden
  float* agg  = carve((size_t)N * FDIM);   // aggregation / next-layer input
  float* asrc = carve((size_t)N * HH);
  float* adst = carve((size_t)N * HH);
  float* mbuf = carve((size_t)N * HH);
  float* zbuf = carve((size_t)N * HH);

  const int BT = 256;
  dim3 blk(BT);
  dim3 g_gemm((N + 127) / 128);                 // 8 waves x 16 rows per block
  dim3 g_nh((N * HH + BT - 1) / BT);
  dim3 g_edge((E + BT - 1) / BT);
  dim3 g_scat(((size_t)E * 32 + BT - 1) / BT);  // one wave per edge
  dim3 g_nf(((size_t)N * FDIM + BT - 1) / BT);

  // ---------------- GAT layer 0 ----------------
  gemm_wmma_f32<FDIM, FDIM><<<g_gemm, blk, 0, stream>>>(x, W0, hbuf, N,
                                                        nullptr, 0);
  att_coef_kernel<<<g_nh, blk, 0, stream>>>(hbuf, att_src0, att_dst0, asrc,
                                            adst, N * HH);
  hipMemsetAsync(mbuf, 0xFF, (size_t)N * HH * sizeof(float), stream);
  hipMemsetAsync(zbuf, 0, (size_t)N * HH * sizeof(float), stream);
  hipMemsetAsync(agg, 0, (size_t)N * FDIM * sizeof(float), stream);
  edge_max_kernel<<<g_edge, blk, 0, stream>>>(src, dst, asrc, adst, mbuf, E);
  edge_sum_kernel<<<g_edge, blk, 0, stream>>>(src, dst, asrc, adst, mbuf, zbuf, E);
  edge_scatter_kernel<<<g_scat, blk, 0, stream>>>(src, dst, asrc, adst, mbuf,
                                                  zbuf, hbuf, agg, E);
  bias_relu_kernel<<<g_nf, blk, 0, stream>>>(agg, b0, FDIM - 1,
                                             (long)N * FDIM);

  // ---------------- GAT layer 1 ----------------
  gemm_wmma_f32<FDIM, FDIM><<<g_gemm, blk, 0, stream>>>(agg, W1, hbuf, N,
                                                        nullptr, 0);
  att_coef_kernel<<<g_nh, blk, 0, stream>>>(hbuf, att_src1, att_dst1, asrc,
                                            adst, N * HH);
  hipMemsetAsync(mbuf, 0xFF, (size_t)N * HH * sizeof(float), stream);
  hipMemsetAsync(zbuf, 0, (size_t)N * HH * sizeof(float), stream);
  hipMemsetAsync(agg, 0, (size_t)N * FDIM * sizeof(float), stream);
  edge_max_kernel<<<g_edge, blk, 0, stream>>>(src, dst, asrc, adst, mbuf, E);
  edge_sum_kernel<<<g_edge, blk, 0, stream>>>(src, dst, asrc, adst, mbuf, zbuf, E);
  edge_scatter_kernel<<<g_scat, blk, 0, stream>>>(src, dst, asrc, adst, mbuf,
                                                  zbuf, hbuf, agg, E);
  bias_relu_kernel<<<g_nf, blk, 0, stream>>>(agg, b1, FDIM - 1,
                                             (long)N * FDIM);

  // ---------------- MLP (bias/ReLU fused into GEMM epilogue) ----------------
  gemm_wmma_f32<FDIM, 32><<<g_gemm, blk, 0, stream>>>(agg, Wm1, hbuf, N,
                                                      bm1, 1);
  gemm_wmma_f32<32, 16><<<g_gemm, blk, 0, stream>>>(hbuf, Wm2, out, N,
                                                    bm2, 0);
}